// RelationalLinear_6536940224889
// MI455X (gfx1250) — compile-verified
//
#include <hip/hip_runtime.h>
#include <hip/hip_bf16.h>

typedef __attribute__((ext_vector_type(2))) float v2f;
typedef __attribute__((ext_vector_type(8))) float v8f;

#define E_TOTAL 524288
#define IN_DIM  256
#define OUT_DIM 256
#define NTYPES  16
#define EPB     256   // edges per block in the gather kernel

// ---------------------------------------------------------------------------
// Kernel 1: y[t] = x[t] @ W[t] for t in 0..15 via V_WMMA_F32_16X16X4_F32.
// One wave (32 threads) per (type t, 16-wide output tile n0).
// A-operand (16x4 f32, per ISA layout): lanes 0-15 -> VGPR0=K0, VGPR1=K1;
// lanes 16-31 -> VGPR0=K2, VGPR1=K3. We broadcast x[t, k] across all M rows,
// so every row of the 16x16 D tile equals y[t, n0:n0+16].
// B-operand (4x16 f32) loaded with the layout symmetric to A (lane-half picks
// the K pair, VGPR index picks the element within the pair).
// D layout: VGPR0 lanes 0-15 = (M=0, N=lane) -> write y from c[0], lanes 0-15.
// ---------------------------------------------------------------------------
__global__ __launch_bounds__(32) void rellin_matvec_wmma(
    const float* __restrict__ x,   // [E, IN]; only rows 0..15 used
    const float* __restrict__ W,   // [T, IN, OUT]
    float* __restrict__ y)         // [T, OUT] (workspace)
{
    const int t    = blockIdx.x >> 4;         // type 0..15
    const int n0   = (blockIdx.x & 15) << 4;  // output tile base 0..240
    const int lane = threadIdx.x;             // 0..31
    const int h    = lane >> 4;               // half-wave: selects K pair
    const int n    = lane & 15;               // column within tile

    const float* __restrict__ xr = x + (size_t)t * IN_DIM;
    const float* __restrict__ Wt = W + (size_t)t * IN_DIM * OUT_DIM;

    v8f c = {};
    #pragma unroll 8
    for (int k0 = 0; k0 < IN_DIM; k0 += 4) {
        const int kk = k0 + (h << 1);
        v2f a, b;
        a.x = xr[kk];                          // A[m, kk]   (broadcast over m)
        a.y = xr[kk + 1];                      // A[m, kk+1]
        b.x = Wt[(size_t)kk * OUT_DIM + n0 + n];       // B[kk,   n]
        b.y = Wt[(size_t)(kk + 1) * OUT_DIM + n0 + n]; // B[kk+1, n]
        // 8 args: (neg_a, A, neg_b, B, c_mod, C, reuse_a, reuse_b)
        c = __builtin_amdgcn_wmma_f32_16x16x4_f32(
                false, a, false, b, (short)0, c, false, false);
    }

    if (lane < 16)
        y[(size_t)t * OUT_DIM + n0 + lane] = c[0];
}

// ---------------------------------------------------------------------------
// Kernel 2: out[e, :] = y[edge_types[e], :].  Pure store-bandwidth problem
// (512 MB out). Stage y (16 KB) + this block's 256 edge types in LDS, then
// emit contiguous 128-bit stores: each inner iteration covers a 4 KB stripe
// across the 256 threads (perfect coalescing).
// ---------------------------------------------------------------------------
__global__ __launch_bounds__(256) void rellin_gather(
    const float4* __restrict__ y4,    // [T * OUT/4] = 1024 float4
    const int*    __restrict__ types, // [E]
    float4*       __restrict__ out4)  // [E * OUT/4]
{
    __shared__ float4 s_y[NTYPES * (OUT_DIM / 4)]; // 16 KB
    __shared__ int    s_t[EPB];                    // 1 KB

    const int tid = threadIdx.x;
    const int e0  = blockIdx.x * EPB;

    // gfx1250 global_prefetch_b8 on the index stream
    __builtin_prefetch(&types[e0 + tid], 0, 0);

    #pragma unroll
    for (int i = 0; i < 4; ++i)
        s_y[tid + i * 256] = y4[tid + i * 256];
    s_t[tid] = types[e0 + tid];
    __syncthreads();

    const size_t base = (size_t)e0 * (OUT_DIM / 4);
    #pragma unroll 4
    for (int i = tid; i < EPB * (OUT_DIM / 4); i += 256) {
        const int e = i >> 6;       // local edge
        const int c = i & 63;       // float4 column
        out4[base + i] = s_y[s_t[e] * (OUT_DIM / 4) + c];
    }
}

extern "C" void kernel_launch(void* const* d_in, const int* in_sizes, int n_in,
                              void* d_out, int out_size, void* d_ws, size_t ws_size,
                              hipStream_t stream) {
    const float* x     = (const float*)d_in[0];   // x_j        [E, IN] f32
    const float* W     = (const float*)d_in[1];   // rel_weights[T, IN, OUT] f32
    const int*   types = (const int*)d_in[2];     // edge_types [E] int32
    float*       out   = (float*)d_out;           // [E, OUT] f32
    float*       y     = (float*)d_ws;            // [T, OUT] f32 scratch (16 KB)

    // 16 types x 16 output tiles, one wave each
    rellin_matvec_wmma<<<NTYPES * (OUT_DIM / 16), 32, 0, stream>>>(x, W, y);

    // bandwidth-bound gather: 2048 blocks x 256 edges
    rellin_gather<<<E_TOTAL / EPB, 256, 0, stream>>>(
        (const float4*)y, types, (float4*)out);
}